// MAGNN_metapath_instance_6889127542845
// MI455X (gfx1250) — compile-verified
//
#include <hip/hip_runtime.h>
#include <math.h>

// MAGNN metapath-instance encoder ('average' rnn_type), MI455X / gfx1250.
//
// Roofline: ~1.07 GB of traffic, 23.3 TB/s HBM -> ~46 us floor; no shared-
// operand contraction exists (each edge averages its OWN 3 gathered rows),
// so WMMA is inapplicable. The CDNA5 win is the async data-movement path:
//   - global_load_async_to_lds_b128 (ASYNCcnt) stages the 3 gathered 2KB rows
//     th:TH_LOAD_NT_RT -> don't thrash WGP$ with random gathers, keep L2 RT
//     (features = 204.8 MB vs 192 MB L2, ~3.9x row reuse -> L2 does the work)
//   - global_store_async_from_lds_b128 th:TH_STORE_NT streams the write-once
//     256 MB output past L2 so it can't evict the feature working set.

#define MP_LEN     4
#define DIM        512
#define NUM_HEADS  8
#define OUT_DIM    64
#define NEPS       1e-12f
#define THREADS    64          // == OUT_DIM; 2 waves of wave32

__global__ __launch_bounds__(THREADS)
void magnn_metapath_avg_kernel(const float* __restrict__ features,
                               const int*   __restrict__ indices,
                               float*       __restrict__ out,
                               int          n_edges)
{
    __shared__ float smem[3 * DIM];          // 6 KB: 3 gathered rows, then reused
                                             // (rows 0..511) for the output row

    const int e = blockIdx.x;
    if (e >= n_edges) return;
    const int t = threadIdx.x;               // coordinate d within a head

    // ---- stage the 3 gathered feature rows into LDS asynchronously ----
    // Per row: 64 lanes x 2 chunks x 16 B = 2048 B (one full contiguous row).
    #pragma unroll
    for (int j = 0; j < MP_LEN - 1; ++j) {
        const int row = indices[e * MP_LEN + j];               // uniform -> s_load
        const float* src = features + (size_t)row * DIM;
        #pragma unroll
        for (int c = 0; c < 2; ++c) {
            const int elt = (t + THREADS * c) * 4;             // float index of 16B chunk
            // low 32 bits of a generic pointer to __shared__ == LDS offset
            unsigned lds_off = (unsigned)(size_t)(&smem[j * DIM + elt]);
            const float* gptr = src + elt;
            asm volatile(
                "global_load_async_to_lds_b128 %0, %1, off th:TH_LOAD_NT_RT"
                :
                : "v"(lds_off), "v"(gptr)
                : "memory");
        }
    }

#if __has_builtin(__builtin_amdgcn_s_wait_asynccnt)
    __builtin_amdgcn_s_wait_asynccnt(0);
#else
    asm volatile("s_wait_asynccnt 0" ::: "memory");
#endif
    __syncthreads();   // both waves' async LDS writes visible to all lanes

    // ---- mean over the 3 metapath positions; thread-local cross-head norm ----
    float acc[NUM_HEADS];
    float sumsq = 0.0f;
    #pragma unroll
    for (int h = 0; h < NUM_HEADS; ++h) {
        const int p = h * OUT_DIM + t;
        const float v = (smem[p] + smem[DIM + p] + smem[2 * DIM + p]) * (1.0f / 3.0f);
        acc[h] = v;
        sumsq += v * v;
    }

    // F.normalize(p=2, dim=head): eft = hidden / max(||hidden||_2, eps)
    const float inv = 1.0f / fmaxf(sqrtf(sumsq), NEPS);

    __syncthreads();   // all lanes finished reading rows before we overwrite

    // ---- assemble the normalized output row in LDS ----
    #pragma unroll
    for (int h = 0; h < NUM_HEADS; ++h)
        smem[h * OUT_DIM + t] = acc[h] * inv;   // bank-conflict-free ds_store

    __syncthreads();   // dscnt drained + cross-wave visibility for the async store path

    // ---- stream the 2 KB row out via async LDS->global, NT past L2 ----
    float* dst = out + (size_t)e * DIM;
    #pragma unroll
    for (int c = 0; c < 2; ++c) {
        const int elt = (t + THREADS * c) * 4;
        unsigned lds_off = (unsigned)(size_t)(&smem[elt]);
        float* gptr = dst + elt;
        asm volatile(
            "global_store_async_from_lds_b128 %0, %1, off th:TH_STORE_NT"
            :
            : "v"(gptr), "v"(lds_off)
            : "memory");
    }
    // S_ENDPGM performs an implicit wait-idle: ASYNCcnt drains before retire.
}

extern "C" void kernel_launch(void* const* d_in, const int* in_sizes, int n_in,
                              void* d_out, int out_size, void* d_ws, size_t ws_size,
                              hipStream_t stream)
{
    const float* features = (const float*)d_in[0];   // (N_NODES, 512) fp32
    const int*   indices  = (const int*)d_in[1];     // (E, 4) int
    float*       out      = (float*)d_out;           // (E, 512) fp32

    const int n_edges = in_sizes[1] / MP_LEN;

    magnn_metapath_avg_kernel<<<n_edges, THREADS, 0, stream>>>(
        features, indices, out, n_edges);
}